// LinearAttention_5995774345734
// MI455X (gfx1250) — compile-verified
//
#include <hip/hip_runtime.h>

typedef __attribute__((ext_vector_type(2))) float v2f;
typedef __attribute__((ext_vector_type(8))) float v8f;
typedef __attribute__((ext_vector_type(4))) unsigned int u32x4;
typedef __attribute__((ext_vector_type(8))) unsigned int u32x8;

constexpr int N_ = 4, L_ = 4096, H_ = 8, D_ = 64;
constexpr int HEADS = N_ * H_;
constexpr int PITCH = 68;   // 64 + 4: rows stay 16B-aligned, column reads bank-conflict-free

__device__ __forceinline__ float featmap(float x) {
    // elu(x) + 1  ==  x+1 (x>0), exp(x) (x<=0)
    return x > 0.f ? x + 1.f : __expf(x);
}

// Tensor Data Mover: load a 64x64 fp32 tile (row stride = row_stride_elems) from
// global memory into LDS, padding each 64-dword row with 4 dwords -> LDS pitch 68.
// D# per CDNA5 ISA (group0: count/lds/global/type, group1: dims).
__device__ __forceinline__ void tdm_load_64x64(const float* gptr, const float* lds_dst,
                                               unsigned row_stride_elems)
{
    unsigned long long ga = (unsigned long long)(size_t)gptr;
    unsigned lds_addr = (unsigned)(size_t)lds_dst;        // low 32 bits = LDS byte offset

    u32x4 g0;
    g0.x = 1u;                                            // count = 1 (valid descriptor)
    g0.y = lds_addr;                                      // lds_addr [63:32]
    g0.z = (unsigned)ga;                                  // global_addr lo
    g0.w = (unsigned)(ga >> 32) | (2u << 30);             // global_addr hi | type=2

    u32x8 g1;
    g1[0] = (2u << 16)      // data_size = 4B
          | (1u << 20)      // pad_enable
          | (5u << 22)      // pad_interval: every 64 dwords
          | (3u << 25);     // pad_amount: 4 dwords
    g1[1] = 64u << 16;      // tensor_dim0 = 64 (low 16 bits at [63:48])
    g1[2] = 64u << 16;      // tensor_dim0 hi = 0 | tensor_dim1 = 64 (low 16 at [95:80])
    g1[3] = 64u << 16;      // tensor_dim1 hi = 0 | tile_dim0 = 64
    g1[4] = 64u;            // tile_dim1 = 64, tile_dim2 = 0 (unused -> 2D)
    g1[5] = row_stride_elems; // tensor_dim0_stride (lo 32, data_size units)
    g1[6] = 0u;
    g1[7] = 0u;

    u32x4 gz = {0u, 0u, 0u, 0u};                          // groups 2/3: unused dims = 0

    asm volatile("tensor_load_to_lds %0, %1, %2, %3"
                 :: "s"(g0), "s"(g1), "s"(gz), "s"(gz)
                 : "memory");
}

__global__ void la_zero(float* __restrict__ p, int n) {
    int i = blockIdx.x * blockDim.x + threadIdx.x;
    if (i < n) p[i] = 0.f;
}

// Phase 1: KV[head][m][d] += sum_s V[s][m]*phi(K)[s][d];  Ksum[head][d] += sum_s phi(K)[s][d]
// All tile movement via TDM, double-buffered so DMA of chunk c+1 overlaps WMMA of chunk c.
// phi(K) is applied in place in LDS, fused with the Ksum reduction pass.
__global__ __launch_bounds__(512) void la_phase1(const float* __restrict__ K,
                                                 const float* __restrict__ V,
                                                 float* __restrict__ KV,
                                                 float* __restrict__ Ksum)
{
    __shared__ float vs[2][64][PITCH];
    __shared__ float ks[2][64][PITCH];
    __shared__ float ksum_s[64];

    const int head = blockIdx.x;
    const int n = head / H_, h = head % H_;
    const int s_base = blockIdx.y * 256;
    const int tid  = threadIdx.x;
    const int wave = tid >> 5, lane = tid & 31;
    const int m0   = (wave >> 2) << 4;
    const int d0   = (wave &  3) << 4;
    const int klo  = (lane >> 4) << 1;
    const int lidx = lane & 15;

    if (tid < 64) ksum_s[tid] = 0.f;

    const size_t rowsz = (size_t)H_ * D_;                 // elements between s rows
    const float* kbase = K + ((size_t)n * L_ + s_base) * rowsz + (size_t)h * D_;
    const float* vbase = V + ((size_t)n * L_ + s_base) * rowsz + (size_t)h * D_;

    if (wave == 0) {                                      // prologue: chunk 0 in flight
        tdm_load_64x64(kbase, &ks[0][0][0], (unsigned)rowsz);
        tdm_load_64x64(vbase, &vs[0][0][0], (unsigned)rowsz);
    }

    v8f acc = {};
    for (int c = 0; c < 4; ++c) {
        const int cur = c & 1, nxt = cur ^ 1;
        if (c) __syncthreads();                           // prev compute done before nxt overwrite
        if (wave == 0) {
            if (c + 1 < 4) {
                tdm_load_64x64(kbase + (size_t)(c + 1) * 64 * rowsz, &ks[nxt][0][0], (unsigned)rowsz);
                tdm_load_64x64(vbase + (size_t)(c + 1) * 64 * rowsz, &vs[nxt][0][0], (unsigned)rowsz);
                __builtin_amdgcn_s_wait_tensorcnt(2);     // chunk c landed; c+1 still in flight
            } else {
                __builtin_amdgcn_s_wait_tensorcnt(0);
            }
        }
        __syncthreads();

        {   // Fused pass: phi(K) in place + Ksum partials. 8 threads per column, 8 rows each.
            const int d  = tid & 63;
            const int sg = (tid >> 6) << 3;
            float part = 0.f;
            #pragma unroll
            for (int j = 0; j < 8; ++j) {
                float p = featmap(ks[cur][sg + j][d]);
                ks[cur][sg + j][d] = p;
                part += p;
            }
            atomicAdd(&ksum_s[d], part);
        }
        __syncthreads();                                  // transformed K visible to all waves

        #pragma unroll
        for (int s = 0; s < 64; s += 4) {
            v2f a, b;
            a.x = vs[cur][s + klo    ][m0 + lidx];        // A = V^T : (m, k)
            a.y = vs[cur][s + klo + 1][m0 + lidx];
            b.x = ks[cur][s + klo    ][d0 + lidx];        // B = phi(K) : (k, d)
            b.y = ks[cur][s + klo + 1][d0 + lidx];
            acc = __builtin_amdgcn_wmma_f32_16x16x4_f32(false, a, false, b,
                                                        (short)0, acc, false, false);
        }
    }
    __syncthreads();

    // Flush: C layout -> M = j + 8*(lane>>4), Ncol = lane&15
    const int cm = m0 + ((lane >> 4) << 3);
    const int cd = d0 + lidx;
    float* kvp = KV + ((size_t)head * 64 + cm) * 64 + cd;
    #pragma unroll
    for (int j = 0; j < 8; ++j) atomicAdd(kvp + (size_t)j * 64, acc[j]);
    if (tid < 64) atomicAdd(&Ksum[head * 64 + tid], ksum_s[tid]);
}

// Phase 2: O[l][m] = z[l] * sum_d phi(Q)[l][d] * KV[m][d]; Q and KV tiles via TDM.
// phi(Q) applied in place in LDS once, then z-dot and WMMA read clean values.
__global__ __launch_bounds__(512) void la_phase2(const float* __restrict__ Q,
                                                 const float* __restrict__ KV,
                                                 const float* __restrict__ Ksum,
                                                 float* __restrict__ Out)
{
    __shared__ float qs[64][PITCH];
    __shared__ float kvs[64][PITCH];
    __shared__ float zs[64];
    __shared__ float ksl[64];

    const int head = blockIdx.x;
    const int n = head / H_, h = head % H_;
    const int l_base = blockIdx.y * 64;
    const int tid  = threadIdx.x;
    const int wave = tid >> 5, lane = tid & 31;
    const int l0   = (wave >> 2) << 4;
    const int m0   = (wave &  3) << 4;
    const int klo  = (lane >> 4) << 1;
    const int lidx = lane & 15;

    if (wave == 0) {
        const float* qbase = Q + ((size_t)(n * L_ + l_base) * H_ + h) * D_;
        tdm_load_64x64(qbase, &qs[0][0], (unsigned)(H_ * D_));
        tdm_load_64x64(KV + (size_t)head * 64 * 64, &kvs[0][0], 64u);
    }
    if (tid < 64) ksl[tid] = Ksum[head * 64 + tid];
    if (wave == 0) __builtin_amdgcn_s_wait_tensorcnt(0);
    __syncthreads();

    {   // phi(Q) in place: 8 threads per column, 8 rows each (conflict-free)
        const int d  = tid & 63;
        const int sg = (tid >> 6) << 3;
        #pragma unroll
        for (int j = 0; j < 8; ++j) qs[sg + j][d] = featmap(qs[sg + j][d]);
    }
    __syncthreads();

    if (tid < 64) {
        float dot = 0.f;
        #pragma unroll
        for (int d = 0; d < 64; ++d) dot += qs[tid][d] * ksl[d];
        zs[tid] = 1.f / (dot + 1e-6f);
    }
    __syncthreads();

    v8f acc = {};
    #pragma unroll
    for (int d = 0; d < 64; d += 4) {
        v2f a, b;
        a.x = qs[l0 + lidx][d + klo];                     // A = phi(Q) : (l, k=d)
        a.y = qs[l0 + lidx][d + klo + 1];
        b.x = kvs[m0 + lidx][d + klo];                    // B = KV^T : (k=d, m)
        b.y = kvs[m0 + lidx][d + klo + 1];
        acc = __builtin_amdgcn_wmma_f32_16x16x4_f32(false, a, false, b,
                                                    (short)0, acc, false, false);
    }

    const int rowoff = (lane >> 4) << 3;
    #pragma unroll
    for (int j = 0; j < 8; ++j) {
        const int cl = l0 + j + rowoff;
        Out[((size_t)(n * L_ + l_base + cl) * H_ + h) * D_ + m0 + lidx] = acc[j] * zs[cl];
    }
}

extern "C" void kernel_launch(void* const* d_in, const int* in_sizes, int n_in,
                              void* d_out, int out_size, void* d_ws, size_t ws_size,
                              hipStream_t stream) {
    const float* q = (const float*)d_in[0];
    const float* k = (const float*)d_in[1];
    const float* v = (const float*)d_in[2];
    float* out  = (float*)d_out;

    float* kv   = (float*)d_ws;                       // HEADS*64*64 floats
    float* ksum = kv + (size_t)HEADS * 64 * 64;       // HEADS*64 floats
    const int ztot = HEADS * (64 * 64 + 64);

    la_zero<<<(ztot + 255) / 256, 256, 0, stream>>>((float*)d_ws, ztot);
    la_phase1<<<dim3(HEADS, L_ / 256), 512, 0, stream>>>(k, v, kv, ksum);
    la_phase2<<<dim3(HEADS, L_ / 64), 512, 0, stream>>>(q, kv, ksum, out);
}